// SparseExtractionPointNet_38903813767583
// MI455X (gfx1250) — compile-verified
//
#include <hip/hip_runtime.h>

typedef __attribute__((ext_vector_type(2))) float v2f;
typedef __attribute__((ext_vector_type(4))) float v4f;
typedef __attribute__((ext_vector_type(8))) float v8f;

#define N_VOX      400000
#define C_PP       128
#define C_SP       256
#define K_DIM      384          // C_PP + C_SP
#define C_OUTD     128
#define PLANE      65536        // H*W floats per (b,c) plane
#define ROWS       64           // rows (voxels) per block
#define XS_STRIDE  388          // 384 + 4 pad -> conflict-free A frag b64 loads
#define WP_STRIDE  288          // dwords per K-pair row: 2*128 + 32 pad (288%64==32)
#define EPSV       1e-5f

#define XS_FLOATS (ROWS * XS_STRIDE)        // 24832
#define WL_FLOATS ((K_DIM / 2) * WP_STRIDE) // 55296
#define ST_FLOATS 256
#define RB_WORDS  ROWS
#define LDS_BYTES ((XS_FLOATS + WL_FLOATS + ST_FLOATS + RB_WORDS) * 4)  // 321792 <= 320KB

__global__ __launch_bounds__(128) void fused_gather_gemm(
    const int* __restrict__ coords, const float* __restrict__ spatial,
    const float* __restrict__ vf, const float* __restrict__ Wg,
    const float* __restrict__ bg, float* __restrict__ hout,
    float* __restrict__ stats_g)
{
  extern __shared__ float smem[];
  float*    xs = smem;                       // x tile: 64 x 384 (stride 388)
  float*    wl = smem + XS_FLOATS;           // W, K-pair-major swizzle
  float*    st = wl + WL_FLOATS;             // per-channel sum / sumsq
  unsigned* rb = (unsigned*)(st + ST_FLOATS);// per-row gather base (float index)

  const int tid = threadIdx.x;
  const int n0  = blockIdx.x * ROWS;

  if (tid < 128) { st[tid] = 0.0f; st[tid + 128] = 0.0f; }
  if (tid < ROWS) {
    const int n  = n0 + tid;
    const int b  = coords[n * 4 + 0];
    const int yy = coords[n * 4 + 2];
    const int xx = coords[n * 4 + 3];
    rb[tid] = (unsigned)(b * (C_SP * PLANE) + yy * 256 + xx);
  }
  __syncthreads();

  // Stage W into LDS, K-pair-major: wl[(k/2)*288 + 2*col + (k&1)] = W[k,col].
  // Global side stays coalesced; B fragments become single aligned b64 loads.
  for (int i = tid; i < K_DIM * C_OUTD; i += 128) {
    const int k = i / C_OUTD, col = i % C_OUTD;
    wl[(k >> 1) * WP_STRIDE + col * 2 + (k & 1)] = Wg[i];
  }
  // Stage voxel features (coalesced float4)
  for (int i = tid; i < (ROWS * C_PP) / 4; i += 128) {
    const int e = i * 4;
    const int r = e / C_PP, c = e % C_PP;
    *(v4f*)(xs + r * XS_STRIDE + c) = *(const v4f*)(vf + (size_t)(n0 + r) * C_PP + c);
  }
  // Gather spatial features: xs[r][128+c] = spatial[b, c, y, x]
  for (int i = tid; i < ROWS * C_SP; i += 128) {
    const int r = i & (ROWS - 1);
    const int c = i >> 6;
    xs[r * XS_STRIDE + C_PP + c] = spatial[(size_t)rb[r] + (size_t)c * PLANE];
  }
  __syncthreads();

  const int wave  = tid >> 5;
  const int lane  = tid & 31;
  const int lrow  = lane & 15;   // A: row in tile / B,D: column in tile
  const int khalf = lane >> 4;   // 0: K={k,k+1}  1: K={k+2,k+3}

  // Accumulators init with bias (every element of a lane is one channel)
  v8f acc[8];
  #pragma unroll
  for (int nt = 0; nt < 8; ++nt) {
    const float bv = bg[nt * 16 + lrow];
    v8f t = { bv, bv, bv, bv, bv, bv, bv, bv };
    acc[nt] = t;
  }

  const float* arow  = xs + (wave * 16 + lrow) * XS_STRIDE + 2 * khalf;
  const float* bbase = wl + khalf * WP_STRIDE + 2 * lrow;  // pair row k/2 + khalf

  for (int k = 0; k < K_DIM; k += 4) {
    const v2f a = *(const v2f*)(arow + k);                 // ds_load_b64
    const float* bk = bbase + (k >> 1) * WP_STRIDE;
    #pragma unroll
    for (int nt = 0; nt < 8; ++nt) {
      const v2f b = *(const v2f*)(bk + nt * 32);           // ds_load_b64, even pair
      acc[nt] = __builtin_amdgcn_wmma_f32_16x16x4_f32(
          false, a, false, b, (short)0, acc[nt], false, false);
    }
  }

  // Epilogue: store pre-BN h, accumulate per-channel sum & sumsq
  const int row0 = n0 + wave * 16 + khalf * 8;             // D layout: M = v + 8*khalf
  #pragma unroll
  for (int nt = 0; nt < 8; ++nt) {
    const int ch = nt * 16 + lrow;
    float s = 0.0f, s2 = 0.0f;
    #pragma unroll
    for (int v = 0; v < 8; ++v) {
      const float val = acc[nt][v];
      hout[(size_t)(row0 + v) * C_OUTD + ch] = val;
      s  += val;
      s2 += val * val;
    }
    atomicAdd(&st[ch], s);
    atomicAdd(&st[128 + ch], s2);
  }
  __syncthreads();
  if (tid < 128) {
    atomicAdd(&stats_g[tid],       st[tid]);
    atomicAdd(&stats_g[128 + tid], st[128 + tid]);
  }
}

__global__ void zero_stats(float* __restrict__ p) { p[threadIdx.x] = 0.0f; }

__global__ void bn_finalize(const float* __restrict__ stats,
                            const float* __restrict__ gamma,
                            const float* __restrict__ beta,
                            float* __restrict__ sc)
{
  const int c = threadIdx.x;                 // 128 threads
  const float inv_n = 1.0f / (float)N_VOX;
  const float mean  = stats[c] * inv_n;
  const float var   = stats[128 + c] * inv_n - mean * mean;
  const float s     = rsqrtf(var + EPSV) * gamma[c];
  sc[c]       = s;
  sc[128 + c] = beta[c] - mean * s;
}

__global__ __launch_bounds__(256) void bn_relu(float* __restrict__ h,
                                               const float* __restrict__ sc)
{
  const size_t i = ((size_t)blockIdx.x * 256 + threadIdx.x) * 4;
  if (i >= (size_t)N_VOX * C_OUTD) return;
  const int ch = (int)(i & (C_OUTD - 1));
  v4f x = *(v4f*)(h + i);
  const v4f s = *(const v4f*)(sc + ch);
  const v4f t = *(const v4f*)(sc + 128 + ch);
  x = x * s + t;
  x.x = fmaxf(x.x, 0.0f);
  x.y = fmaxf(x.y, 0.0f);
  x.z = fmaxf(x.z, 0.0f);
  x.w = fmaxf(x.w, 0.0f);
  *(v4f*)(h + i) = x;
}

extern "C" void kernel_launch(void* const* d_in, const int* in_sizes, int n_in,
                              void* d_out, int out_size, void* d_ws, size_t ws_size,
                              hipStream_t stream) {
  (void)in_sizes; (void)n_in; (void)out_size; (void)ws_size;
  const int*   coords  = (const int*)d_in[0];    // (N,4) int32
  const float* spatial = (const float*)d_in[1];  // (4,256,256,256) f32
  const float* vf      = (const float*)d_in[2];  // (N,128) f32
  const float* Wg      = (const float*)d_in[3];  // (384,128) f32
  const float* bg      = (const float*)d_in[4];  // (128,) f32
  const float* gamma   = (const float*)d_in[5];  // (128,) f32
  const float* beta    = (const float*)d_in[6];  // (128,) f32
  float* out   = (float*)d_out;                  // (N,128) f32
  float* stats = (float*)d_ws;                   // [0..127]=sum [128..255]=sumsq
  float* sc    = stats + 256;                    // [0..127]=scale [128..255]=shift

  // Allow >64KB dynamic LDS (gfx1250 WGP has 320KB).
  (void)hipFuncSetAttribute((const void*)fused_gather_gemm,
                            hipFuncAttributeMaxDynamicSharedMemorySize, LDS_BYTES);

  zero_stats<<<1, 256, 0, stream>>>(stats);
  fused_gather_gemm<<<N_VOX / ROWS, 128, LDS_BYTES, stream>>>(
      coords, spatial, vf, Wg, bg, out, stats);
  bn_finalize<<<1, 128, 0, stream>>>(stats, gamma, beta, sc);
  const int total4 = (N_VOX * C_OUTD) / 4;
  bn_relu<<<(total4 + 255) / 256, 256, 0, stream>>>(out, sc);
}